// AddMessagePasser_9509057593721
// MI455X (gfx1250) — compile-verified
//
#include <hip/hip_runtime.h>
#include <hip/hip_bf16.h>

typedef __attribute__((ext_vector_type(2))) float v2f;
typedef __attribute__((ext_vector_type(4))) float v4f;
typedef __attribute__((ext_vector_type(8))) float v8f;

#define DDIM     512
#define N_NODES  100000
#define NUM_RELS 500
#define BM       32          // edges per block
#define SLD      516         // padded LDS row stride (floats): 516%64==4 -> conflict-free b64 frags
#define NT       8           // 16-wide N tiles per wave (wave covers 16x128)

__global__ __launch_bounds__(256)
void rgcn_msg_gemm_kernel(const float* __restrict__ node_feat,
                          const float* __restrict__ edge_feat,
                          const int*   __restrict__ src_idx,
                          const int*   __restrict__ edge_type,
                          const float* __restrict__ emb,
                          const float* __restrict__ W,
                          const float* __restrict__ bias,
                          float* __restrict__ out,
                          int E)
{
    __shared__ float Alds[BM * SLD];   // 66048 B of the 320 KB WGP LDS

    const int tid   = threadIdx.x;
    const int lane  = tid & 31;
    const int wave  = tid >> 5;            // 0..7
    const int waveM = wave >> 2;           // 0..1  (16-row slab)
    const int waveN = wave & 3;            // 0..3  (128-col slab)
    const int rowBase = blockIdx.x * BM;

    // ---------------- Stage 1: gather msg rows into LDS ----------------
    // msg[r] = emb[edge_type[e]] + node_feat[src_idx[e]] + edge_feat[e]
    // 8 threads per row, float4-vectorized, interleaved for coalescing.
    // node_feat/emb: cached (node table ~205MB vs 192MB L2 -> keep resident).
    // edge_feat: streamed once -> non-temporal, don't evict node_feat.
    {
        const int row  = tid >> 3;         // 0..31
        const int cseg = tid & 7;          // 0..7
        int e  = rowBase + row;
        int eC = (e < E) ? e : (E - 1);
        int src = src_idx[eC];
        int et  = edge_type[eC];
        if ((unsigned)src >= (unsigned)N_NODES) src = 0;
        if ((unsigned)et  >= (unsigned)NUM_RELS) et = 0;
        const v4f* __restrict__ nP = (const v4f*)(node_feat + (size_t)src * DDIM);
        const v4f* __restrict__ rP = (const v4f*)(emb       + (size_t)et  * DDIM);
        const v4f* __restrict__ eP = (const v4f*)(edge_feat + (size_t)eC  * DDIM);
        float* dstRow = &Alds[row * SLD];
#pragma unroll
        for (int i = 0; i < 16; ++i) {
            int c4 = cseg + i * 8;                     // 0..127
            v4f v = rP[c4] + nP[c4] + __builtin_nontemporal_load(eP + c4);
            *(v4f*)(dstRow + c4 * 4) = v;
        }
    }
    __syncthreads();

    // ---------------- Stage 2: K-loop of f32 WMMA ----------------
    // A frag (16x4):  lane L holds A[L%16][k + 2*(L/16) + r]   -> ds_load_b64
    // B frag (4x16):  B[k][n] = W[n][k]; lane L reads W[n0+L%16][k+2*(L/16)+r] -> global_load_b64
    v8f acc[NT];
#pragma unroll
    for (int t = 0; t < NT; ++t) acc[t] = (v8f){0.f,0.f,0.f,0.f,0.f,0.f,0.f,0.f};

    const int mloc  = (lane & 15);
    const int kHalf = (lane >> 4) << 1;                 // 0 or 2
    const float* __restrict__ Arow  = &Alds[(waveM * 16 + mloc) * SLD + kHalf];
    const float* __restrict__ Wlane = W + (size_t)(waveN * 128 + mloc) * DDIM + kHalf;

#pragma unroll 2
    for (int k = 0; k < DDIM; k += 4) {
        v2f a = *(const v2f*)(Arow + k);
#pragma unroll
        for (int t = 0; t < NT; ++t) {
            v2f bf = *(const v2f*)(Wlane + (size_t)(t * 16) * DDIM + k);
            acc[t] = __builtin_amdgcn_wmma_f32_16x16x4_f32(
                /*neg_a*/false, a, /*neg_b*/false, bf,
                /*c_mod*/(short)0, acc[t], /*reuse_a*/false, /*reuse_b*/false);
        }
    }

    // ---------------- Stage 3: bias + ReLU + store ----------------
    // C/D layout: VGPR j, lane L -> row M = j + 8*(L/16), col N = L%16
    // Output is write-once -> non-temporal stores (keep L2 for node_feat).
    const int mHi  = (lane >> 4) << 3;                  // 0 or 8
    const int eW   = rowBase + waveM * 16 + mHi;        // first edge row of this half-wave
    if (rowBase + BM <= E) {
        // Fast path (4882 of 4883 blocks): no per-element predication.
#pragma unroll
        for (int t = 0; t < NT; ++t) {
            const int f = waveN * 128 + t * 16 + mloc;
            const float bv = bias[f];
            float* __restrict__ op = out + (size_t)eW * DDIM + f;
#pragma unroll
            for (int j = 0; j < 8; ++j) {
                float v = acc[t][j] + bv;
                v = v > 0.f ? v : 0.f;
                __builtin_nontemporal_store(v, op + (size_t)j * DDIM);
            }
        }
    } else {
        // Tail block: guard each row.
#pragma unroll
        for (int t = 0; t < NT; ++t) {
            const int f = waveN * 128 + t * 16 + mloc;
            const float bv = bias[f];
#pragma unroll
            for (int j = 0; j < 8; ++j) {
                int e = eW + j;
                float v = acc[t][j] + bv;
                v = v > 0.f ? v : 0.f;
                if (e < E) __builtin_nontemporal_store(v, out + (size_t)e * DDIM + f);
            }
        }
    }
}

extern "C" void kernel_launch(void* const* d_in, const int* in_sizes, int n_in,
                              void* d_out, int out_size, void* d_ws, size_t ws_size,
                              hipStream_t stream) {
    (void)n_in; (void)out_size; (void)d_ws; (void)ws_size;
    const float* node_feat = (const float*)d_in[0];
    const float* edge_feat = (const float*)d_in[1];
    const int*   src_idx   = (const int*)  d_in[2];
    const int*   edge_type = (const int*)  d_in[3];
    const float* emb       = (const float*)d_in[4];
    const float* W         = (const float*)d_in[5];
    const float* bias      = (const float*)d_in[6];
    float*       out       = (float*)d_out;

    const int E = in_sizes[2];                 // 156250
    const int grid = (E + BM - 1) / BM;        // 4883
    rgcn_msg_gemm_kernel<<<grid, 256, 0, stream>>>(
        node_feat, edge_feat, src_idx, edge_type, emb, W, bias, out, E);
}